// SoftAttention_17205638988695
// MI455X (gfx1250) — compile-verified
//
#include <hip/hip_runtime.h>
#include <hip/hip_bf16.h>

typedef __attribute__((ext_vector_type(2))) float v2f;
typedef __attribute__((ext_vector_type(8))) float v8f;

#define B_LEN 16
#define S_LEN 2048
#define H_LEN 1024
#define E_LEN 2048
#define NCHUNK 32          // S-chunks per batch (64 rows each)
#define ROWS   64          // rows per chunk

// ---------------------------------------------------------------------------
// C[16 x N] = A[16 x K] * B[K x N]   using V_WMMA_F32_16X16X4_F32 (wave32).
// One wave per block; block x covers a 16-wide N tile.
// TRANSB=1: B element (k,n) at Bm[n*K + k] (torch [out,in]); else Bm[k*N + n].
// Fragment layouts per CDNA5 ISA 7.12.2.
// ---------------------------------------------------------------------------
template <int TRANSB>
__global__ void gemm16_wmma(const float* __restrict__ A, const float* __restrict__ Bm,
                            float* __restrict__ C, int K, int N) {
    const int lane = threadIdx.x & 31;
    const int l15  = lane & 15;
    const int half = lane >> 4;
    const int kb   = half * 2;
    const int n0   = blockIdx.x * 16;

    v8f acc = {};
    #pragma unroll 4
    for (int k0 = 0; k0 < K; k0 += 4) {
        v2f a;
        const float* ap = A + (size_t)l15 * K + k0 + kb;
        a.x = ap[0];
        a.y = ap[1];

        v2f b;
        if (TRANSB) {
            const float* bp = Bm + (size_t)(n0 + l15) * K + k0 + kb;
            b.x = bp[0];
            b.y = bp[1];
        } else {
            b.x = Bm[(size_t)(k0 + kb + 0) * N + n0 + l15];
            b.y = Bm[(size_t)(k0 + kb + 1) * N + n0 + l15];
        }
        acc = __builtin_amdgcn_wmma_f32_16x16x4_f32(false, a, false, b,
                                                    (short)0, acc, false, false);
    }

    const int mbase = half * 8;
    #pragma unroll
    for (int v = 0; v < 8; ++v) {
        C[(size_t)(mbase + v) * N + n0 + l15] = acc[v];
    }
}

// ---------------------------------------------------------------------------
// Fused: scores + local softmax stats + partial context for one 64-row chunk.
// Single HBM pass over `value`: the context re-read of the same 512 KB chunk
// hits L2 (192 MB).  Grid: (NCHUNK, B), 256 threads (8 waves).
// Outputs: raw scores (for final alphas), m_arr/sum_arr (chunk stats),
//          pctx (locally-normalized partial context).
// ---------------------------------------------------------------------------
__global__ void attn_part_kernel(const float* __restrict__ value,
                                 const float* __restrict__ veff,
                                 const int* __restrict__ mask,
                                 float* __restrict__ scores,
                                 float* __restrict__ m_arr,
                                 float* __restrict__ sum_arr,
                                 float* __restrict__ pctx) {
    __shared__ float sveff[E_LEN];
    __shared__ float sc[ROWS];
    __shared__ float wexp[ROWS];

    const int b    = blockIdx.y;
    const int z    = blockIdx.x;
    const int tid  = threadIdx.x;
    const int wave = tid >> 5;
    const int lane = tid & 31;
    const int s0   = z * ROWS;

    for (int i = tid; i < E_LEN; i += 256) sveff[i] = veff[b * E_LEN + i];
    __syncthreads();

    const float* vb = value + (size_t)b * S_LEN * E_LEN;

    // ---- phase 1: scores for 64 rows (one row per wave per iteration) ----
    for (int r = wave; r < ROWS; r += 8) {
        const int s = s0 + r;
        const float* row = vb + (size_t)s * E_LEN;
        __builtin_prefetch(row + (size_t)8 * E_LEN + lane * 4, 0, 0);

        float acc = 0.f;
        #pragma unroll
        for (int c = 0; c < E_LEN; c += 128) {
            const float4 v4 = *(const float4*)(row + c + lane * 4);
            const int e = c + lane * 4;
            acc += v4.x * sveff[e + 0] + v4.y * sveff[e + 1]
                 + v4.z * sveff[e + 2] + v4.w * sveff[e + 3];
        }
        #pragma unroll
        for (int off = 16; off > 0; off >>= 1)
            acc += __shfl_down(acc, off, 32);
        if (lane == 0) {
            const float scv = (mask[b * S_LEN + s] == 0) ? -__builtin_inff() : acc;
            sc[r] = scv;
            scores[b * S_LEN + s] = scv;
        }
    }
    __syncthreads();

    // ---- phase 2: local softmax stats over the 64 chunk scores (wave 0) ----
    if (tid < 32) {
        const float a0 = sc[tid];
        const float a1 = sc[tid + 32];
        float m = fmaxf(a0, a1);
        #pragma unroll
        for (int off = 16; off > 0; off >>= 1)
            m = fmaxf(m, __shfl_down(m, off, 32));
        m = __shfl(m, 0, 32);
        const bool dead = (m == -__builtin_inff());   // fully-masked chunk
        const float w0 = dead ? 0.f : __expf(a0 - m);
        const float w1 = dead ? 0.f : __expf(a1 - m);
        wexp[tid]      = w0;
        wexp[tid + 32] = w1;
        float s = w0 + w1;
        #pragma unroll
        for (int off = 16; off > 0; off >>= 1)
            s += __shfl_down(s, off, 32);
        if (tid == 0) {
            m_arr[b * NCHUNK + z]   = m;
            sum_arr[b * NCHUNK + z] = s;
        }
    }
    __syncthreads();

    // ---- phase 3: partial context, re-reading the chunk (L2-resident) ----
    float acc[8] = {};
    const float* base = vb + (size_t)s0 * E_LEN + tid;
    for (int j = 0; j < ROWS; ++j) {
        const float w = wexp[j];
        const float* rp = base + (size_t)j * E_LEN;
        #pragma unroll
        for (int i = 0; i < 8; ++i)
            acc[i] = fmaf(w, rp[i * 256], acc[i]);
    }
    float* pc = pctx + (size_t)(b * NCHUNK + z) * E_LEN + tid;
    #pragma unroll
    for (int i = 0; i < 8; ++i)
        pc[i * 256] = acc[i];
}

// ---------------------------------------------------------------------------
// Global stats per batch: gm[b] = max_z m_z ; gs[b] = sum_z sum_z*exp(m_z-gm).
// One wave per batch.
// ---------------------------------------------------------------------------
__global__ void stats_combine(const float* __restrict__ m_arr,
                              const float* __restrict__ sum_arr,
                              float* __restrict__ gm, float* __restrict__ gs) {
    const int b    = blockIdx.x;
    const int lane = threadIdx.x;
    const float m  = m_arr[b * NCHUNK + lane];
    float mr = m;
    #pragma unroll
    for (int off = 16; off > 0; off >>= 1)
        mr = fmaxf(mr, __shfl_down(mr, off, 32));
    mr = __shfl(mr, 0, 32);
    const float scale = (m == -__builtin_inff()) ? 0.f : __expf(m - mr);
    float s = sum_arr[b * NCHUNK + lane] * scale;
    #pragma unroll
    for (int off = 16; off > 0; off >>= 1)
        s += __shfl_down(s, off, 32);
    if (lane == 0) { gm[b] = mr; gs[b] = s; }
}

// ---------------------------------------------------------------------------
// context[b,e] = ( sum_z pctx[b,z,e] * exp(m_z - gm[b]) ) / gs[b]
// Grid: (E/256, B), 256 threads.
// ---------------------------------------------------------------------------
__global__ void ctx_final(const float* __restrict__ pctx,
                          const float* __restrict__ m_arr,
                          const float* __restrict__ gm,
                          const float* __restrict__ gs,
                          float* __restrict__ ctx) {
    __shared__ float sscale[NCHUNK];
    const int b = blockIdx.y;
    const int e = blockIdx.x * 256 + threadIdx.x;
    if (threadIdx.x < NCHUNK) {
        const float m = m_arr[b * NCHUNK + threadIdx.x];
        sscale[threadIdx.x] = (m == -__builtin_inff()) ? 0.f : __expf(m - gm[b]);
    }
    __syncthreads();
    float acc = 0.f;
    #pragma unroll 4
    for (int z = 0; z < NCHUNK; ++z)
        acc = fmaf(sscale[z], pctx[(size_t)(b * NCHUNK + z) * E_LEN + e], acc);
    ctx[b * E_LEN + e] = acc / gs[b];
}

// ---------------------------------------------------------------------------
// alphas[b,s] = exp(scores[b,s] - gm[b]) / gs[b]
// ---------------------------------------------------------------------------
__global__ void alphas_final(const float* __restrict__ scores,
                             const float* __restrict__ gm,
                             const float* __restrict__ gs,
                             float* __restrict__ alphas) {
    const int i = blockIdx.x * 256 + threadIdx.x;   // 0 .. B*S-1
    const int b = i / S_LEN;
    alphas[i] = __expf(scores[i] - gm[b]) / gs[b];
}

// ---------------------------------------------------------------------------
extern "C" void kernel_launch(void* const* d_in, const int* in_sizes, int n_in,
                              void* d_out, int out_size, void* d_ws, size_t ws_size,
                              hipStream_t stream) {
    const float* query = (const float*)d_in[0];   // [16,1,1024]
    const float* value = (const float*)d_in[1];   // [16,2048,2048]
    const int*   mask  = (const int*)d_in[2];     // [16,1,2048]
    const float* W_enc = (const float*)d_in[3];   // [1024,2048]
    const float* W_q   = (const float*)d_in[4];   // [1024,1024]
    const float* W_bil = (const float*)d_in[5];   // [1024,1024]

    float* out    = (float*)d_out;
    float* ctx    = out;                          // [16,2048]
    float* alphas = out + B_LEN * E_LEN;          // [16,2048]

    float* ws      = (float*)d_ws;
    float* qproj   = ws;                          // 16*1024
    float* qW      = ws + 16384;                  // 16*1024
    float* veff    = ws + 32768;                  // 16*2048
    float* scores  = ws + 65536;                  // 16*2048
    float* m_arr   = ws + 98304;                  // 16*32
    float* sum_arr = ws + 98816;                  // 16*32
    float* gm      = ws + 99328;                  // 16
    float* gs      = ws + 99344;                  // 16
    float* pctx    = ws + 131072;                 // 32*16*2048 (4 MB)

    // 1) q_proj = query @ W_q^T
    gemm16_wmma<1><<<H_LEN / 16, 32, 0, stream>>>(query, W_q, qproj, H_LEN, H_LEN);
    // 2) qW = q_proj @ W_bil
    gemm16_wmma<0><<<H_LEN / 16, 32, 0, stream>>>(qproj, W_bil, qW, H_LEN, H_LEN);
    // 3) v_eff = qW @ W_enc
    gemm16_wmma<0><<<E_LEN / 16, 32, 0, stream>>>(qW, W_enc, veff, H_LEN, E_LEN);
    // 4) fused scores + local softmax + partial context (single HBM pass)
    attn_part_kernel<<<dim3(NCHUNK, B_LEN), 256, 0, stream>>>(
        value, veff, mask, scores, m_arr, sum_arr, pctx);
    // 5) global softmax stats
    stats_combine<<<B_LEN, 32, 0, stream>>>(m_arr, sum_arr, gm, gs);
    // 6) combine partial contexts -> context output
    ctx_final<<<dim3(E_LEN / 256, B_LEN), 256, 0, stream>>>(pctx, m_arr, gm, gs, ctx);
    // 7) final alphas from saved raw scores
    alphas_final<<<B_LEN * S_LEN / 256, 256, 0, stream>>>(scores, gm, gs, alphas);
}